// SyntheticMatchingLoss_45715631899327
// MI455X (gfx1250) — compile-verified
//
#include <hip/hip_runtime.h>
#include <hip/hip_bf16.h>

typedef __attribute__((ext_vector_type(2))) float v2f;
typedef __attribute__((ext_vector_type(8))) float v8f;

#define NVOX   8000    // 20*20*20
#define NCH    16
#define NTILES 500     // 8000 / 16

// ---------------------------------------------------------------------------
// Pass 1: antialiased triangle resize along H and W (64x64 -> 20x20), keep Z=64.
// Matches jax.image.resize(method='trilinear', antialias=True): triangle kernel
// of half-width scale=3.2, per-dim weights normalized over in-range taps.
// Layout in/out: (b, c, z, h, w). blockIdx.y selects source(0)/target(1).
// ---------------------------------------------------------------------------
__global__ void smloss_resize_hw(const float* __restrict__ src0,
                                 const float* __restrict__ src1,
                                 float* __restrict__ tmp0,
                                 float* __restrict__ tmp1) {
  int idx = blockIdx.x * blockDim.x + threadIdx.x;
  const int total = 2 * 16 * 64 * 20 * 20;   // 819200
  if (idx >= total) return;
  const float* src = blockIdx.y ? src1 : src0;
  float*       dst = blockIdx.y ? tmp1 : tmp0;

  int w  = idx % 20;
  int h  = (idx / 20) % 20;
  int z  = (idx / 400) % 64;
  int bc = idx / 25600;                       // b*16 + c  (0..31)

  const float* plane = src + ((size_t)bc * 64 + (size_t)z) * 4096; // 64*64
  const float S = 3.2f;
  float fh = (h + 0.5f) * S - 0.5f;
  float fw = (w + 0.5f) * S - 0.5f;
  int h0 = (int)ceilf(fh - S);  if (h0 < 0)  h0 = 0;
  int h1 = (int)floorf(fh + S); if (h1 > 63) h1 = 63;
  int w0 = (int)ceilf(fw - S);  if (w0 < 0)  w0 = 0;
  int w1 = (int)floorf(fw + S); if (w1 > 63) w1 = 63;

  float wxs = 0.f;
  for (int ix = w0; ix <= w1; ++ix) {
    float wx = S - fabsf((float)ix - fw); if (wx < 0.f) wx = 0.f;
    wxs += wx;
  }
  float acc = 0.f, wys = 0.f;
  for (int iy = h0; iy <= h1; ++iy) {
    float wy = S - fabsf((float)iy - fh); if (wy < 0.f) wy = 0.f;
    const float* prow = plane + iy * 64;
    float racc = 0.f;
    for (int ix = w0; ix <= w1; ++ix) {
      float wx = S - fabsf((float)ix - fw); if (wx < 0.f) wx = 0.f;
      racc = fmaf(wx, prow[ix], racc);
    }
    acc = fmaf(wy, racc, acc);
    wys += wy;
  }
  dst[idx] = acc / (wxs * wys);
}

// ---------------------------------------------------------------------------
// Pass 2: resize along D (64 -> 20) and transpose into feature layout
// feat[(b*N + n)*16 + c], n = d*400 + h*20 + w  (matches reshape+transpose).
// ---------------------------------------------------------------------------
__global__ void smloss_resize_d(const float* __restrict__ tmp0,
                                const float* __restrict__ tmp1,
                                float* __restrict__ s_feat,
                                float* __restrict__ t_feat) {
  int idx = blockIdx.x * blockDim.x + threadIdx.x;
  const int total = 2 * NVOX * NCH;           // 256000
  if (idx >= total) return;
  const float* tmp = blockIdx.y ? tmp1 : tmp0;
  float*      feat = blockIdx.y ? t_feat : s_feat;

  int c  = idx % 16;
  int t  = idx / 16;                          // b*N + n
  int n  = t % NVOX;
  int b  = t / NVOX;
  int d  = n / 400;
  int hw = n % 400;

  const float S = 3.2f;
  float fd = (d + 0.5f) * S - 0.5f;
  int d0 = (int)ceilf(fd - S);  if (d0 < 0)  d0 = 0;
  int d1 = (int)floorf(fd + S); if (d1 > 63) d1 = 63;

  const float* base = tmp + (size_t)(b * 16 + c) * 25600 + hw;
  float acc = 0.f, wzs = 0.f;
  for (int iz = d0; iz <= d1; ++iz) {
    float wz = S - fabsf((float)iz - fd); if (wz < 0.f) wz = 0.f;
    acc = fmaf(wz, base[iz * 400], acc);
    wzs += wz;
  }
  feat[idx] = acc / wzs;
}

// ---------------------------------------------------------------------------
// ||t_m||^2 per target voxel (argmin only needs  ||t||^2 - 2 s.t ).
// ---------------------------------------------------------------------------
__global__ void smloss_tnorm(const float* __restrict__ t_feat,
                             float* __restrict__ tnorm) {
  int i = blockIdx.x * blockDim.x + threadIdx.x;
  if (i >= 2 * NVOX) return;
  const float* r = t_feat + (size_t)i * NCH;
  float s = 0.f;
#pragma unroll
  for (int k = 0; k < NCH; ++k) s = fmaf(r[k], r[k], s);
  tnorm[i] = s;
}

// ---------------------------------------------------------------------------
// WMMA argmin: one wave32 per (batch, 16-row tile of s). The 16x16 cross tile
// s_tile . t_tile^T is built from 4 chained V_WMMA_F32_16X16X4_F32 (K=16 f32).
// A operand stays resident in VGPRs for the
// whole 500-iteration column loop; t features + tnorm stream from L2/WGP$.
// Per ISA 16x16 f32 C/D layout: lane L holds column L%16, rows (L/16)*8+v.
// ---------------------------------------------------------------------------
__global__ void __launch_bounds__(256)
smloss_argmin(const float* __restrict__ s_feat,
              const float* __restrict__ t_feat,
              const float* __restrict__ tnorm,
              int* __restrict__ nearest) {
  int lane = threadIdx.x & 31;
  int wave = threadIdx.x >> 5;
  int work = blockIdx.x * 8 + wave;           // wave-uniform
  if (work >= 2 * NTILES) return;
  int b  = work / NTILES;
  int nt = work % NTILES;

  int row = lane & 15;
  int hi  = lane >> 4;

  // A operand (s tile): per ISA 16x4 f32 A layout, lane L holds
  // s[row = L%16][k0 + 2*(L/16) + {0,1}] for each K-chunk k0 = 0,4,8,12.
  const float* srow = s_feat + ((size_t)(b * NVOX + nt * 16 + row)) * NCH + 2 * hi;
  v2f a0 = *(const v2f*)(srow + 0);
  v2f a1 = *(const v2f*)(srow + 4);
  v2f a2 = *(const v2f*)(srow + 8);
  v2f a3 = *(const v2f*)(srow + 12);

  float minv[8];
  int   mini[8];
#pragma unroll
  for (int v = 0; v < 8; ++v) { minv[v] = 3.4e38f; mini[v] = 0; }

  const float* tbase = t_feat + (size_t)b * NVOX * NCH;
  const float* tnb   = tnorm + (size_t)b * NVOX;

  for (int mt = 0; mt < NTILES; ++mt) {
    // B operand (t tile, used as K x N): same per-lane addressing pattern.
    const float* trow = tbase + (size_t)(mt * 16 + row) * NCH + 2 * hi;
    v2f b0 = *(const v2f*)(trow + 0);
    v2f b1 = *(const v2f*)(trow + 4);
    v2f b2 = *(const v2f*)(trow + 8);
    v2f b3 = *(const v2f*)(trow + 12);
    float tn = tnb[mt * 16 + row];

    v8f acc = {};
    acc = __builtin_amdgcn_wmma_f32_16x16x4_f32(false, a0, false, b0, (short)0, acc, false, false);
    acc = __builtin_amdgcn_wmma_f32_16x16x4_f32(false, a1, false, b1, (short)0, acc, false, false);
    acc = __builtin_amdgcn_wmma_f32_16x16x4_f32(false, a2, false, b2, (short)0, acc, false, false);
    acc = __builtin_amdgcn_wmma_f32_16x16x4_f32(false, a3, false, b3, (short)0, acc, false, false);

    int col = mt * 16 + row;
#pragma unroll
    for (int v = 0; v < 8; ++v) {
      float score = fmaf(-2.0f, acc[v], tn);  // ||t||^2 - 2 s.t  (||s||^2 const per row)
      if (score < minv[v]) { minv[v] = score; mini[v] = col; }
    }
  }

  // Reduce across the 16 lanes of each half (rows 0-7 in lanes 0-15, rows 8-15
  // in lanes 16-31); smallest-index tie-break == argmin first-occurrence.
#pragma unroll
  for (int m = 1; m < 16; m <<= 1) {
#pragma unroll
    for (int v = 0; v < 8; ++v) {
      float ov = __shfl_xor(minv[v], m, 32);
      int   oi = __shfl_xor(mini[v], m, 32);
      if (ov < minv[v] || (ov == minv[v] && oi < mini[v])) { minv[v] = ov; mini[v] = oi; }
    }
  }
  if (row == 0) {
    int* out = nearest + b * NVOX + nt * 16 + hi * 8;
#pragma unroll
    for (int v = 0; v < 8; ++v) out[v] = mini[v];
  }
}

// ---------------------------------------------------------------------------
// Loss: sum |expected - grid[nearest]| with deterministic two-stage reduction.
// grid[m] = (m/400, (m/20)%20, m%20)  (meshgrid 'ij' of (D,H,W)).
// ---------------------------------------------------------------------------
__global__ void smloss_partial(const int* __restrict__ nearest,
                               const float* __restrict__ expected,
                               float* __restrict__ partials) {
  __shared__ float sm[256];
  int i = blockIdx.x * blockDim.x + threadIdx.x;  // over B*N
  float s = 0.f;
  if (i < 2 * NVOX) {
    int m = nearest[i];
    float gd = (float)(m / 400);
    float gh = (float)((m / 20) % 20);
    float gw = (float)(m % 20);
    const float* e = expected + (size_t)i * 3;
    s = fabsf(e[0] - gd) + fabsf(e[1] - gh) + fabsf(e[2] - gw);
  }
  sm[threadIdx.x] = s;
  __syncthreads();
  for (int st = 128; st > 0; st >>= 1) {
    if (threadIdx.x < st) sm[threadIdx.x] += sm[threadIdx.x + st];
    __syncthreads();
  }
  if (threadIdx.x == 0) partials[blockIdx.x] = sm[0];
}

__global__ void smloss_final(const float* __restrict__ partials,
                             float* __restrict__ out, int nparts) {
  __shared__ float sm[128];
  float s = 0.f;
  for (int i = threadIdx.x; i < nparts; i += 128) s += partials[i];
  sm[threadIdx.x] = s;
  __syncthreads();
  for (int st = 64; st > 0; st >>= 1) {
    if (threadIdx.x < st) sm[threadIdx.x] += sm[threadIdx.x + st];
    __syncthreads();
  }
  if (threadIdx.x == 0) out[0] = sm[0] * (1.0f / (2.0f * NVOX * 3.0f));
}

// ---------------------------------------------------------------------------
extern "C" void kernel_launch(void* const* d_in, const int* in_sizes, int n_in,
                              void* d_out, int out_size, void* d_ws, size_t ws_size,
                              hipStream_t stream) {
  const float* src = (const float*)d_in[0];   // (2,16,64,64,64)
  const float* tgt = (const float*)d_in[1];   // (2,16,64,64,64)
  const float* exp = (const float*)d_in[2];   // (2,8000,3)
  float* out = (float*)d_out;

  // Workspace layout (floats): ~8.7 MB total
  float* ws       = (float*)d_ws;
  float* tmp0     = ws;                        // 819200  (b,c,64,20,20) source
  float* tmp1     = ws + 819200;               // 819200  target
  float* s_feat   = ws + 1638400;              // 256000  (b,n,16)
  float* t_feat   = ws + 1894400;              // 256000
  float* tnormp   = ws + 2150400;              // 16000
  int*   nearest  = (int*)(ws + 2166400);      // 16000
  float* partials = ws + 2182400;              // 64

  smloss_resize_hw<<<dim3(3200, 2), 256, 0, stream>>>(src, tgt, tmp0, tmp1);
  smloss_resize_d <<<dim3(1000, 2), 256, 0, stream>>>(tmp0, tmp1, s_feat, t_feat);
  smloss_tnorm    <<<63, 256, 0, stream>>>(t_feat, tnormp);
  smloss_argmin   <<<125, 256, 0, stream>>>(s_feat, t_feat, tnormp, nearest);
  smloss_partial  <<<63, 256, 0, stream>>>(nearest, exp, partials);
  smloss_final    <<<1, 128, 0, stream>>>(partials, out, 63);
}